// SSIM_48052094107907
// MI455X (gfx1250) — compile-verified
//
#include <hip/hip_runtime.h>

typedef __attribute__((ext_vector_type(2))) float v2f;
typedef __attribute__((ext_vector_type(8))) float v8f;

#define OUTD 502
#define IMGD 512
#define NB   16
#define NC   3
#define TILE 64
#define PATCH 74                      // TILE + 10 halo
#define LSTR 80                       // padded LDS row stride (floats)
#define BLKS_PER_IMG   64             // 8x8 tiles of 64x64 outputs
#define BLKS_PER_BATCH (NC * BLKS_PER_IMG)   // 192
#define NBLOCKS        (NB * BLKS_PER_BATCH) // 3072

__global__ __launch_bounds__(256) void ssim_main(const float* __restrict__ X,
                                                 const float* __restrict__ Y,
                                                 float* __restrict__ part)
{
    __shared__ float ldsX[PATCH * LSTR];   // 23.1 KB
    __shared__ float ldsY[PATCH * LSTR];   // 23.1 KB
    __shared__ float ldsSv[8 * 16 * 28];   // per-wave vertical sums, 14 KB
    __shared__ float waveAcc[8];

    const int tid = threadIdx.x;
    const int bid = blockIdx.x;
    const int b   = bid / BLKS_PER_BATCH;
    const int r0m = bid % BLKS_PER_BATCH;
    const int ch  = r0m / BLKS_PER_IMG;
    const int tt  = r0m % BLKS_PER_IMG;
    const int row0 = (tt >> 3) * TILE;
    const int col0 = (tt & 7)  * TILE;

    const size_t base = ((size_t)(b * NC + ch)) * (size_t)(IMGD * IMGD);

    // ---- Stage patch (with clip to [0,1]) into LDS, coalesced ----
    for (int i = tid; i < PATCH * PATCH; i += 256) {
        int pr = i / PATCH, pc = i % PATCH;
        int gr = row0 + pr, gc = col0 + pc;
        float xv = 0.f, yv = 0.f;
        if (gr < IMGD && gc < IMGD) {
            size_t o = base + (size_t)gr * IMGD + gc;
            xv = fminf(fmaxf(X[o], 0.f), 1.f);
            yv = fminf(fmaxf(Y[o], 0.f), 1.f);
        }
        ldsX[pr * LSTR + pc] = xv;
        ldsY[pr * LSTR + pc] = yv;
    }
    __syncthreads();

    const int lane = tid & 31;
    const int wid  = tid >> 5;
    const int m    = lane & 15;   // A-row / B-col / D-col index
    const int hi   = lane >> 4;   // lane half
    const int kb   = hi * 2;      // K sub-index base within a K=4 chunk
    float* Sv = &ldsSv[wid * 16 * 28];

    const float inv = 1.0f / 121.0f;
    const float C1 = 1e-4f, C2 = 9e-4f;
    float accum = 0.f;

    // ---- A operand: banded ones, depends only on (k, lane): hoist once ----
    v2f aband[7];
    #pragma unroll
    for (int k = 0; k < 7; ++k) {
        const int kg0 = 4 * k + kb;
        const int kg1 = kg0 + 1;
        aband[k].x = (kg0 >= m && kg0 <= m + 10) ? 1.0f : 0.0f;
        aband[k].y = (kg1 >= m && kg1 <= m + 10) ? 1.0f : 0.0f;
    }

    for (int sub = 0; sub < 2; ++sub) {
        const int st  = wid * 2 + sub;        // 16 subtiles of 16x16 in 64x64
        const int sr0 = (st >> 2) * 16;
        const int sc0 = (st & 3)  * 16;

        // ---- Preload B-operand raw x/y values once per subtile (registers
        //      survive the s_wait_dscnt memory clobbers; avoids 5x LDS reload)
        v2f bxv[2][7], byv[2][7];
        #pragma unroll
        for (int h = 0; h < 2; ++h) {
            const int cc = sc0 + 16 * h + m;
            #pragma unroll
            for (int k = 0; k < 7; ++k) {
                const int kg0 = 4 * k + kb;
                const int kg1 = kg0 + 1;
                // clamp address; aband is zero for kg > 25 so value is ignored
                const int rr0 = sr0 + (kg0 > 25 ? 25 : kg0);
                const int rr1 = sr0 + (kg1 > 25 ? 25 : kg1);
                v2f bx, by;
                bx.x = ldsX[rr0 * LSTR + cc];
                bx.y = ldsX[rr1 * LSTR + cc];
                by.x = ldsY[rr0 * LSTR + cc];
                by.y = ldsY[rr1 * LSTR + cc];
                bxv[h][k] = bx;
                byv[h][k] = by;
            }
        }

        float S[5][8];                         // horizontal sums per map
        #pragma unroll
        for (int mp = 0; mp < 5; ++mp) {       // maps: x, y, x^2, y^2, xy
            #pragma unroll
            for (int h = 0; h < 2; ++h) {      // column halves (cols 0-15, 16-25)
                v8f acc = {};
                #pragma unroll
                for (int k = 0; k < 7; ++k) {  // K = 26 (padded to 28), chunks of 4
                    const v2f bx = bxv[h][k];
                    const v2f by = byv[h][k];
                    v2f bb;
                    if      (mp == 0) bb = bx;
                    else if (mp == 1) bb = by;
                    else if (mp == 2) bb = bx * bx;
                    else if (mp == 3) bb = by * by;
                    else              bb = bx * by;
                    acc = __builtin_amdgcn_wmma_f32_16x16x4_f32(
                        false, aband[k], false, bb, (short)0, acc, false, false);
                }
                // D layout: lane holds col m; VGPR v holds row v + 8*hi
                #pragma unroll
                for (int v = 0; v < 8; ++v) {
                    const int rrow = v + 8 * hi;
                    const int ncol = 16 * h + m;
                    if (ncol < 26) Sv[rrow * 28 + ncol] = acc[v];
                }
            }
            asm volatile("s_wait_dscnt 0" ::: "memory");
            // Horizontal 11-tap rolling sum: lane -> (row m, col group hi)
            float tv[18];
            #pragma unroll
            for (int i = 0; i < 18; ++i) tv[i] = Sv[m * 28 + 8 * hi + i];
            float sacc = 0.f;
            #pragma unroll
            for (int i = 0; i < 11; ++i) sacc += tv[i];
            S[mp][0] = sacc;
            #pragma unroll
            for (int j = 1; j < 8; ++j) { sacc += tv[j + 10] - tv[j - 1]; S[mp][j] = sacc; }
            asm volatile("s_wait_dscnt 0" ::: "memory");  // WAR before next map's stores
        }

        // ---- SSIM formula + masked accumulate ----
        #pragma unroll
        for (int j = 0; j < 8; ++j) {
            float mu1 = S[0][j] * inv, mu2 = S[1][j] * inv;
            float mu11 = mu1 * mu1, mu22 = mu2 * mu2, mu12 = mu1 * mu2;
            float v1  = S[2][j] * inv - mu11;
            float v2  = S[3][j] * inv - mu22;
            float v12 = S[4][j] * inv - mu12;
            float num = (2.f * mu12 + C1) * (2.f * v12 + C2);
            float den = (mu11 + mu22 + C1) * (v1 + v2 + C2);
            const int gr = row0 + sr0 + m;
            const int gc = col0 + sc0 + 8 * hi + j;
            if (gr < OUTD && gc < OUTD) accum += num / den;
        }
    }

    // ---- Deterministic reduction: wave shuffle -> fixed-order LDS sum ----
    #pragma unroll
    for (int off = 16; off >= 1; off >>= 1)
        accum += __shfl_xor(accum, off, 32);
    if (lane == 0) waveAcc[wid] = accum;
    __syncthreads();
    if (tid == 0) {
        float s = 0.f;
        #pragma unroll
        for (int w = 0; w < 8; ++w) s += waveAcc[w];
        part[bid] = s;
    }
}

__global__ __launch_bounds__(64) void ssim_reduce(const float* __restrict__ part,
                                                  float* __restrict__ out)
{
    __shared__ float s[64];
    const int b = blockIdx.x, t = threadIdx.x;
    float v = 0.f;
    for (int i = t; i < BLKS_PER_BATCH; i += 64)   // fixed order, deterministic
        v += part[b * BLKS_PER_BATCH + i];
    s[t] = v;
    __syncthreads();
    for (int off = 32; off >= 1; off >>= 1) {
        if (t < off) s[t] += s[t + off];
        __syncthreads();
    }
    if (t == 0) out[b] = s[0] * (1.0f / (NC * (float)OUTD * (float)OUTD));
}

extern "C" void kernel_launch(void* const* d_in, const int* in_sizes, int n_in,
                              void* d_out, int out_size, void* d_ws, size_t ws_size,
                              hipStream_t stream)
{
    const float* x = (const float*)d_in[0];
    const float* y = (const float*)d_in[1];
    float* part = (float*)d_ws;                     // 3072 floats of scratch
    ssim_main<<<NBLOCKS, 256, 0, stream>>>(x, y, part);
    ssim_reduce<<<NB, 64, 0, stream>>>(part, (float*)d_out);
}